// QuantumLayer_65481071404620
// MI455X (gfx1250) — compile-verified
//
#include <hip/hip_runtime.h>

// QuantumLayer reference collapses analytically to out[i] = cos(theta[i]):
//   4x Ry(theta) product state -> CNOT ladder (permutation, commutes with Z0)
//   -> <Z0> = cos^2(theta/2) - sin^2(theta/2) = cos(theta).
// Memory-bound: 8 B/elem traffic, ~1.44 us floor at 23.3 TB/s on MI455X.
// Strategy: B128 vectorized streaming loads/stores + v_cos_f32, wave32 blocks.

__global__ __launch_bounds__(256) void quantum_expz_cos_kernel(
    const float* __restrict__ theta, float* __restrict__ out, int n) {
  const int n4 = n >> 2;                       // number of float4 vectors
  const int stride = gridDim.x * blockDim.x;
  const float4* __restrict__ t4 = reinterpret_cast<const float4*>(theta);
  float4* __restrict__ o4 = reinterpret_cast<float4*>(out);

  for (int i = blockIdx.x * blockDim.x + threadIdx.x; i < n4; i += stride) {
    float4 t = t4[i];                          // global_load_b128
    float4 r;
    r.x = cosf(t.x);
    r.y = cosf(t.y);
    r.z = cosf(t.z);
    r.w = cosf(t.w);
    o4[i] = r;                                 // global_store_b128
  }

  // Scalar tail for n % 4 != 0 (n = 4194304 is divisible; kept for robustness).
  const int tail_start = n4 << 2;
  const int ti = tail_start + blockIdx.x * blockDim.x + threadIdx.x;
  if (ti < n) {
    out[ti] = cosf(theta[ti]);
  }
}

extern "C" void kernel_launch(void* const* d_in, const int* in_sizes, int n_in,
                              void* d_out, int out_size, void* d_ws, size_t ws_size,
                              hipStream_t stream) {
  (void)n_in; (void)out_size; (void)d_ws; (void)ws_size;
  const float* theta = reinterpret_cast<const float*>(d_in[0]);
  float* out = reinterpret_cast<float*>(d_out);
  const int n = in_sizes[0];                   // 4,194,304

  const int block = 256;                       // 8 wave32 waves per block
  int n4 = n >> 2;
  int grid = (n4 + block - 1) / block;         // 4096 blocks for n = 4M
  if (grid < 1) grid = 1;

  quantum_expz_cos_kernel<<<grid, block, 0, stream>>>(theta, out, n);
}